// vsgnet_72438918414399
// MI455X (gfx1250) — compile-verified
//
#include <hip/hip_runtime.h>

// Problem constants (fixed by the reference).
#define BATCH     256
#define MAX_RELS  512
#define MAX_OBJ   64
#define DDIM      1024
#define V4_PER_ROW (DDIM / 4)   // 256 float4 per row

typedef float v4f __attribute__((ext_vector_type(4)));

// ---------------------------------------------------------------------------
// Kernel 1: exclusive prefix sums of num_rels and num_obj (B == 256) in a
// single 256-thread block. Hillis–Steele scan in LDS; read-barrier-write
// pattern keeps it race-free under wave32.
// ---------------------------------------------------------------------------
__global__ void vsgnet_scan_offsets(const int* __restrict__ num_rels,
                                    const int* __restrict__ num_obj,
                                    int* __restrict__ rel_off,
                                    int* __restrict__ obj_off) {
    __shared__ int s0[BATCH];
    __shared__ int s1[BATCH];
    const int t = threadIdx.x;
    s0[t] = num_rels[t];
    s1[t] = num_obj[t];
    __syncthreads();
    for (int off = 1; off < BATCH; off <<= 1) {
        int a0 = 0, a1 = 0;
        if (t >= off) { a0 = s0[t - off]; a1 = s1[t - off]; }
        __syncthreads();
        s0[t] += a0;
        s1[t] += a1;
        __syncthreads();
    }
    // inclusive -> exclusive
    rel_off[t] = (t == 0) ? 0 : s0[t - 1];
    obj_off[t] = (t == 0) ? 0 : s1[t - 1];
}

// ---------------------------------------------------------------------------
// Kernel 2: one 256-thread block per relation slot (grid = [MAX_RELS, B]).
// Each thread moves one float4 per source row: fully coalesced
// global_load_b128 x2, VALU average, non-temporal global_store_b128 so the
// streamed 256MB output does not evict the 64MB gather table from the
// 192MB L2.
// ---------------------------------------------------------------------------
__global__ void vsgnet_pair_mean_scatter(const float* __restrict__ g,
                                         const int* __restrict__ num_rels,
                                         const int* __restrict__ obj_pairs,
                                         const int* __restrict__ rel_off,
                                         const int* __restrict__ obj_off,
                                         float* __restrict__ out) {
    const int j = blockIdx.x;          // relation slot within frame
    const int b = blockIdx.y;          // frame
    if (j >= num_rels[b]) return;      // invalid slot: one scalar load, exit

    const int  base = obj_off[b];
    const long pidx = ((long)b * MAX_RELS + j) * 2;
    const long i0   = (long)(obj_pairs[pidx + 0] + base);
    const long i1   = (long)(obj_pairs[pidx + 1] + base);
    const long dest = (long)(rel_off[b] + j);

    const v4f* __restrict__ a = (const v4f*)(g + i0 * DDIM);
    const v4f* __restrict__ c = (const v4f*)(g + i1 * DDIM);
    v4f*       __restrict__ o = (v4f*)(out + dest * DDIM);

    const int t = threadIdx.x;         // 256 threads == 256 float4 per row
    v4f va = a[t];                     // global_load_b128 (RT: keep L2-hot)
    v4f vc = c[t];                     // global_load_b128
    v4f r  = 0.5f * (va + vc);
    __builtin_nontemporal_store(r, o + t);  // global_store_b128, TH=NT
}

// ---------------------------------------------------------------------------
// Host launch: all work on `stream`, graph-capture safe (no sync APIs).
//   d_in[0] : graphical_branch_output  float32 [B*MAX_OBJ, D]
//   d_in[1] : num_rels                 int32   [B]
//   d_in[2] : num_obj                  int32   [B]
//   d_in[3] : obj_pairs                int32   [B, MAX_RELS, 2]
//   d_out   : float32 [tot, D]  (every row written exactly once)
//   d_ws    : scratch; we use 2*B ints for the exclusive prefix sums
// ---------------------------------------------------------------------------
extern "C" void kernel_launch(void* const* d_in, const int* in_sizes, int n_in,
                              void* d_out, int out_size, void* d_ws, size_t ws_size,
                              hipStream_t stream) {
    (void)in_sizes; (void)n_in; (void)out_size; (void)ws_size;

    const float* g        = (const float*)d_in[0];
    const int*   num_rels = (const int*)d_in[1];
    const int*   num_obj  = (const int*)d_in[2];
    const int*   pairs    = (const int*)d_in[3];
    float*       out      = (float*)d_out;

    int* rel_off = (int*)d_ws;          // [B]
    int* obj_off = rel_off + BATCH;     // [B]

    vsgnet_scan_offsets<<<1, BATCH, 0, stream>>>(num_rels, num_obj,
                                                 rel_off, obj_off);

    dim3 grid(MAX_RELS, BATCH);
    vsgnet_pair_mean_scatter<<<grid, V4_PER_ROW, 0, stream>>>(
        g, num_rels, pairs, rel_off, obj_off, out);
}